// GCN_38276748541987
// MI455X (gfx1250) — compile-verified
//
#include <hip/hip_runtime.h>

typedef float v2f __attribute__((ext_vector_type(2)));
typedef float v8f __attribute__((ext_vector_type(8)));

#define CDIM 128   // D_IN = C1 = C2 = 128

// ---------------- utility ----------------
__global__ void k_zero(float* __restrict__ p, int n) {
  int i = blockIdx.x * blockDim.x + threadIdx.x;
  if (i < n) p[i] = 0.0f;
}

// deg[dst] += 1 per edge (self-loop added analytically in k_dinv)
__global__ void k_deg(const int* __restrict__ dst, float* __restrict__ deg, int e) {
  int i = blockIdx.x * blockDim.x + threadIdx.x;
  if (i < e) atomicAdd(deg + dst[i], 1.0f);
}

__global__ void k_dinv(float* __restrict__ deg, int n) {
  int i = blockIdx.x * blockDim.x + threadIdx.x;
  if (i < n) deg[i] = rsqrtf(deg[i] + 1.0f);   // deg incl. self loop >= 1
}

// ---------------- fp32 WMMA GEMM: Y[n x 128] = X[n x 128] @ W[128 x 128] ----
// one wave -> one 16x16 tile of Y; K swept in steps of 4 with v_wmma_f32_16x16x4_f32
__global__ void k_gemm_wmma(const float* __restrict__ X, const float* __restrict__ W,
                            float* __restrict__ Y, int n) {
  const int wave = threadIdx.x >> 5;           // 4 waves / block
  const int lane = threadIdx.x & 31;
  const int row0 = blockIdx.x * 16;            // M tile
  const int col0 = (blockIdx.y * 4 + wave) * 16;  // N tile (8 total)

  int mrow = row0 + (lane & 15);
  if (mrow >= n) mrow = n - 1;                 // clamp (per-lane data only, no EXEC change)
  const int khalf = (lane >> 4) * 2;           // 0 for lanes 0-15, 2 for 16-31
  const int ncol  = col0 + (lane & 15);

  const float* __restrict__ xp = X + (size_t)mrow * CDIM;
  v8f acc = {};
  #pragma unroll 4
  for (int k = 0; k < CDIM; k += 4) {
    // A 16x4: lane holds row (lane&15), K = k+khalf, k+khalf+1  (8B aligned)
    v2f a = *(const v2f*)(xp + k + khalf);
    // B 4x16: lane holds col (lane&15); VGPR0 -> K = k+khalf, VGPR1 -> K = k+khalf+1
    v2f b;
    b.x = W[(size_t)(k + khalf) * CDIM + ncol];
    b.y = W[(size_t)(k + khalf + 1) * CDIM + ncol];
    acc = __builtin_amdgcn_wmma_f32_16x16x4_f32(
        /*neg_a=*/false, a, /*neg_b=*/false, b,
        /*c_mod=*/(short)0, acc, /*reuse_a=*/false, /*reuse_b=*/false);
  }
  // C/D layout: VGPR r, lanes 0-15 -> M = r ; lanes 16-31 -> M = r + 8 ; N = lane&15
  const int mbase = row0 + ((lane >> 4) * 8);
  float* __restrict__ yp = Y + (size_t)mbase * CDIM + ncol;
  if (row0 + 16 <= n) {
    // full tile: uniform branch, no per-store exec masking
    #pragma unroll
    for (int r = 0; r < 8; ++r) yp[(size_t)r * CDIM] = acc[r];
  } else {
    #pragma unroll
    for (int r = 0; r < 8; ++r)
      if (mbase + r < n) yp[(size_t)r * CDIM] = acc[r];
  }
}

// AGG[i] = H[i] * dinv[i]^2   (self-loop term; full overwrite = init)
__global__ void k_selfloop(const float* __restrict__ H, const float* __restrict__ dinv,
                           float* __restrict__ AGG, int n) {
  int idx = blockIdx.x * blockDim.x + threadIdx.x;
  if (idx < n * CDIM) {
    int i = idx >> 7;
    float s = dinv[i];
    AGG[idx] = H[idx] * s * s;
  }
}

// one wave per edge; lane handles 4 consecutive columns (float4 gather + 4 atomics)
__global__ void k_edge_scatter(const float* __restrict__ H, float* __restrict__ AGG,
                               const int* __restrict__ src, const int* __restrict__ dst,
                               const float* __restrict__ dinv, int e) {
  int edge = blockIdx.x * (blockDim.x >> 5) + (threadIdx.x >> 5);
  if (edge >= e) return;
  const int lane = threadIdx.x & 31;
  const int s = src[edge], d = dst[edge];
  const float norm = dinv[s] * dinv[d];
  const float4 v = ((const float4*)(H + (size_t)s * CDIM))[lane];
  float* ap = AGG + (size_t)d * CDIM + lane * 4;
  atomicAdd(ap + 0, v.x * norm);
  atomicAdd(ap + 1, v.y * norm);
  atomicAdd(ap + 2, v.z * norm);
  atomicAdd(ap + 3, v.w * norm);
}

__global__ void k_bias_relu(float* __restrict__ H, const float* __restrict__ b, int n) {
  int idx = blockIdx.x * blockDim.x + threadIdx.x;
  if (idx < n * CDIM) H[idx] = fmaxf(H[idx] + b[idx & (CDIM - 1)], 0.0f);
}

// per-column mean/var -> affine scale/shift (deterministic block reduction)
__global__ void k_bn_stats(const float* __restrict__ H, const float* __restrict__ gamma,
                           const float* __restrict__ beta, float* __restrict__ scale,
                           float* __restrict__ shift, int n) {
  const int c = blockIdx.x;            // 0..127
  float s = 0.0f, sq = 0.0f;
  for (int i = threadIdx.x; i < n; i += blockDim.x) {
    float v = H[(size_t)i * CDIM + c];
    s += v; sq += v * v;
  }
  __shared__ float ss[256], ssq[256];
  ss[threadIdx.x] = s; ssq[threadIdx.x] = sq;
  __syncthreads();
  for (int off = 128; off > 0; off >>= 1) {
    if ((int)threadIdx.x < off) {
      ss[threadIdx.x]  += ss[threadIdx.x + off];
      ssq[threadIdx.x] += ssq[threadIdx.x + off];
    }
    __syncthreads();
  }
  if (threadIdx.x == 0) {
    float inv_n = 1.0f / (float)n;
    float mean = ss[0] * inv_n;
    float var  = ssq[0] * inv_n - mean * mean;   // biased, like reference
    float r = rsqrtf(var + 1e-5f) * gamma[c];
    scale[c] = r;
    shift[c] = beta[c] - mean * r;
  }
}

__global__ void k_bn_apply(float* __restrict__ H, const float* __restrict__ scale,
                           const float* __restrict__ shift, int n) {
  int idx = blockIdx.x * blockDim.x + threadIdx.x;
  if (idx < n * CDIM) {
    int c = idx & (CDIM - 1);
    H[idx] = H[idx] * scale[c] + shift[c];
  }
}

// one wave per node; float4 per lane; atomic accumulation into per-graph sums
__global__ void k_pool(const float* __restrict__ H, const int* __restrict__ batch,
                       float* __restrict__ pool, float* __restrict__ cnt, int n) {
  int i = blockIdx.x * (blockDim.x >> 5) + (threadIdx.x >> 5);
  if (i >= n) return;
  const int lane = threadIdx.x & 31;
  const int g = batch[i];
  const float4 v = ((const float4*)(H + (size_t)i * CDIM))[lane];
  float* pp = pool + (size_t)g * CDIM + lane * 4;
  atomicAdd(pp + 0, v.x);
  atomicAdd(pp + 1, v.y);
  atomicAdd(pp + 2, v.z);
  atomicAdd(pp + 3, v.w);
  if (lane == 0) atomicAdd(cnt + g, 1.0f);
}

// out[g][j] = sum_k (pool[g][k]/cnt[g] + b2[k]) * Wl[k][j] + bl[j]
// (bias b2 of layer 2 folded in here since mean(h + b2) = mean(h) + b2)
__global__ void k_final(const float* __restrict__ pool, const float* __restrict__ cnt,
                        const float* __restrict__ b2, const float* __restrict__ Wl,
                        const float* __restrict__ bl, float* __restrict__ out, int g_total) {
  int t = blockIdx.x * blockDim.x + threadIdx.x;
  if (t >= g_total * 2) return;
  const int g = t >> 1, j = t & 1;
  const float inv_c = 1.0f / fmaxf(cnt[g], 1.0f);
  float acc = bl[j];
  for (int k = 0; k < CDIM; ++k)
    acc += (pool[(size_t)g * CDIM + k] * inv_c + b2[k]) * Wl[k * 2 + j];
  out[t] = acc;
}

// ---------------- driver ----------------
extern "C" void kernel_launch(void* const* d_in, const int* in_sizes, int n_in,
                              void* d_out, int out_size, void* d_ws, size_t ws_size,
                              hipStream_t stream) {
  const float* x     = (const float*)d_in[0];
  const int*   edge  = (const int*)d_in[1];
  const int*   batch = (const int*)d_in[2];
  const float* W1    = (const float*)d_in[3];
  const float* b1    = (const float*)d_in[4];
  const float* gamma = (const float*)d_in[5];
  const float* beta  = (const float*)d_in[6];
  const float* W2    = (const float*)d_in[7];
  const float* b2    = (const float*)d_in[8];
  const float* Wl    = (const float*)d_in[9];
  const float* bl    = (const float*)d_in[10];
  float* out = (float*)d_out;

  const int N = in_sizes[2];        // 50000 nodes
  const int E = in_sizes[1] / 2;    // 600000 edges
  const int G = out_size / 2;       // 64 graphs
  const int* src = edge;
  const int* dst = edge + E;

  float* ws    = (float*)d_ws;
  float* bufA  = ws;                              // N*128
  float* bufB  = bufA + (size_t)N * CDIM;         // N*128
  float* dinv  = bufB + (size_t)N * CDIM;         // N  (deg, then deg^-1/2)
  float* scale = dinv + N;                        // 128
  float* shift = scale + CDIM;                    // 128
  float* pool  = shift + CDIM;                    // G*128
  float* cnt   = pool + (size_t)G * CDIM;         // G

  const int nc = N * CDIM;
  const int zn = N + 2 * CDIM + G * CDIM + G;     // dinv..cnt contiguous region

  // degrees -> dinv
  k_zero<<<(zn + 255) / 256, 256, 0, stream>>>(dinv, zn);
  k_deg <<<(E + 255) / 256, 256, 0, stream>>>(dst, dinv, E);
  k_dinv<<<(N + 255) / 256, 256, 0, stream>>>(dinv, N);

  dim3 ggrid((N + 15) / 16, 2);   // 3125 M-tiles x (2 blocks * 4 waves = 8 N-tiles)

  // layer 1: GEMM -> normalize+aggregate -> bias+relu -> batchnorm
  k_gemm_wmma   <<<ggrid, 128, 0, stream>>>(x, W1, bufA, N);
  k_selfloop    <<<(nc + 255) / 256, 256, 0, stream>>>(bufA, dinv, bufB, N);
  k_edge_scatter<<<(E + 7) / 8, 256, 0, stream>>>(bufA, bufB, src, dst, dinv, E);
  k_bias_relu   <<<(nc + 255) / 256, 256, 0, stream>>>(bufB, b1, N);
  k_bn_stats    <<<CDIM, 256, 0, stream>>>(bufB, gamma, beta, scale, shift, N);
  k_bn_apply    <<<(nc + 255) / 256, 256, 0, stream>>>(bufB, scale, shift, N);

  // layer 2: GEMM -> normalize+aggregate (b2 folded into head)
  k_gemm_wmma   <<<ggrid, 128, 0, stream>>>(bufB, W2, bufA, N);
  k_selfloop    <<<(nc + 255) / 256, 256, 0, stream>>>(bufA, dinv, bufB, N);
  k_edge_scatter<<<(E + 7) / 8, 256, 0, stream>>>(bufA, bufB, src, dst, dinv, E);

  // mean-pool + head
  k_pool <<<(N + 7) / 8, 256, 0, stream>>>(bufB, batch, pool, cnt, N);
  k_final<<<1, 128, 0, stream>>>(pool, cnt, b2, Wl, bl, out, G);
}